// GCNEncoder_31714038514067
// MI455X (gfx1250) — compile-verified
//
#include <hip/hip_runtime.h>
#include <stdint.h>

#define N_NODES 50000
#define N_EDGES 640000
#define HDIM    128
#define NLAYERS 8
#define NGRAPH  64
#define LN_EPS  1e-5f

typedef float v2f __attribute__((ext_vector_type(2)));
typedef float v8f __attribute__((ext_vector_type(8)));

// ------------------------------------------------------------------
// Degree / normalization precompute
// ------------------------------------------------------------------
__global__ void deg_init_kernel(float* __restrict__ deg) {
    int i = blockIdx.x * blockDim.x + threadIdx.x;
    if (i < N_NODES) deg[i] = 1.0f;  // self-loop contributes 1
}

__global__ void deg_acc_kernel(const long long* __restrict__ ei,
                               float* __restrict__ deg) {
    int e = blockIdx.x * blockDim.x + threadIdx.x;
    if (e < N_EDGES) {
        int d = (int)ei[N_EDGES + e];  // dst row of edge_index
        atomicAdd(&deg[d], 1.0f);
    }
}

__global__ void dinv_kernel(float* __restrict__ deg) {
    int i = blockIdx.x * blockDim.x + threadIdx.x;
    if (i < N_NODES) {
        float v = deg[i];
        deg[i] = (v > 0.0f) ? rsqrtf(v) : 0.0f;
    }
}

__global__ void enorm_kernel(const long long* __restrict__ ei,
                             const float* __restrict__ dinv,
                             float* __restrict__ enorm) {
    int e = blockIdx.x * blockDim.x + threadIdx.x;
    if (e < N_EDGES + N_NODES) {
        int s, d;
        if (e < N_EDGES) { s = (int)ei[e]; d = (int)ei[N_EDGES + e]; }
        else             { s = d = e - N_EDGES; }
        enorm[e] = dinv[s] * dinv[d];
    }
}

// ------------------------------------------------------------------
// WMMA GEMM: out[r,0:128] = A[r,0:128] @ W[128,128] (+ bias)
// Block = 256 threads = 8 waves; wave computes 16 rows x 128 cols.
// V_WMMA_F32_16X16X4_F32 (fp32 in / fp32 acc), K chunk = 4.
//   A 16x4 (32-bit): lanes 0-15 hold K={0,1}, lanes 16-31 hold K={2,3}
//   B 4x16 (32-bit): mirror of A with N across lanes
// W is staged in LDS packed as K-pairs so every B fragment is ONE
// contiguous ds_load_b64 (no register shuffles in the inner loop):
//   sW2[kp*128 + n] = { W[2*kp][n], W[2*kp+1][n] }
// ------------------------------------------------------------------
__global__ __launch_bounds__(256)
void gemm128_kernel(const float* __restrict__ A,
                    const float* __restrict__ W,
                    const float* __restrict__ bias,
                    float* __restrict__ out, int nrows) {
    __shared__ v2f sW2[(HDIM / 2) * HDIM];  // 64 KB

    int tid = threadIdx.x;
    // Stage + pack W into LDS: 8192 float2, 32 per thread.
#pragma unroll
    for (int i = 0; i < 32; ++i) {
        int idx = tid + 256 * i;          // idx = kp*128 + n
        int kp = idx >> 7;
        int n  = idx & 127;
        v2f w2;
        w2.x = W[(2 * kp)     * HDIM + n];
        w2.y = W[(2 * kp + 1) * HDIM + n];
        sW2[idx] = w2;
    }
    __syncthreads();

    int wave    = tid >> 5;
    int lane    = tid & 31;
    int halfsel = lane >> 4;   // 0: K pair {0,1}, rows M+0 | 1: K pair {2,3}, rows M+8
    int l16     = lane & 15;
    int rowBase = blockIdx.x * 128 + wave * 16;

    int arow = rowBase + l16;
    if (arow >= nrows) arow = nrows - 1;  // clamp (stores are masked)

    v8f acc[8];
#pragma unroll
    for (int nt = 0; nt < 8; ++nt) {
        v8f z = {0.f, 0.f, 0.f, 0.f, 0.f, 0.f, 0.f, 0.f};
        acc[nt] = z;
    }

    const float* Arow  = A + (size_t)arow * HDIM + halfsel * 2;
    const v2f*   Bbase = sW2 + halfsel * HDIM + l16;

#pragma unroll
    for (int k = 0; k < HDIM; k += 4) {
        v2f a = *(const v2f*)(Arow + k);          // one global_load_b64
        const v2f* Bk = Bbase + (k >> 1) * HDIM;  // kp = k/2 + halfsel
#pragma unroll
        for (int nt = 0; nt < 8; ++nt) {
            v2f b = Bk[nt * 16];                  // one ds_load_b64, imm offset
            acc[nt] = __builtin_amdgcn_wmma_f32_16x16x4_f32(
                false, a, false, b, (short)0, acc[nt], false, false);
        }
    }

    // C/D 16x16 f32 layout: VGPR v -> row (v + 8*halfsel), col = l16 within tile
#pragma unroll
    for (int nt = 0; nt < 8; ++nt) {
        int col = nt * 16 + l16;
        float bb = bias ? bias[col] : 0.0f;
#pragma unroll
        for (int v = 0; v < 8; ++v) {
            int r = rowBase + v + 8 * halfsel;
            if (r < nrows) out[(size_t)r * HDIM + col] = acc[nt][v] + bb;
        }
    }
}

// ------------------------------------------------------------------
// agg[n, :] = bias[:]   (pre-fill with GCNConv bias, then scatter-add)
// ------------------------------------------------------------------
__global__ __launch_bounds__(256)
void agg_init_kernel(const float* __restrict__ bias, float* __restrict__ agg) {
    int i = blockIdx.x * blockDim.x + threadIdx.x;  // over N*HDIM/4 float4s
    if (i < N_NODES * (HDIM / 4)) {
        float4 b = ((const float4*)bias)[i & 31];
        ((float4*)agg)[i] = b;
    }
}

// ------------------------------------------------------------------
// Scatter: agg[dst] += m[src] * enorm  — one wave per edge, float4/lane
// ------------------------------------------------------------------
__global__ __launch_bounds__(256)
void scatter_kernel(const long long* __restrict__ ei,
                    const float* __restrict__ enorm,
                    const float* __restrict__ m,
                    float* __restrict__ agg) {
    int wave = threadIdx.x >> 5;
    int lane = threadIdx.x & 31;
    int e = blockIdx.x * 8 + wave;
    if (e >= N_EDGES + N_NODES) return;
    int s, d;
    if (e < N_EDGES) { s = (int)ei[e]; d = (int)ei[N_EDGES + e]; }
    else             { s = d = e - N_EDGES; }
    float w = enorm[e];
    float4 v = ((const float4*)(m + (size_t)s * HDIM))[lane];
    float* dst = agg + (size_t)d * HDIM + lane * 4;
    atomicAdd(dst + 0, v.x * w);
    atomicAdd(dst + 1, v.y * w);
    atomicAdd(dst + 2, v.z * w);
    atomicAdd(dst + 3, v.w * w);
}

// ------------------------------------------------------------------
// LayerNorm + ReLU: one wave32 per node, float4 per lane (128 feats)
// ------------------------------------------------------------------
__global__ __launch_bounds__(256)
void ln_relu_kernel(const float* __restrict__ in,
                    const float* __restrict__ gamma,
                    const float* __restrict__ beta,
                    float* __restrict__ out) {
    int wave = threadIdx.x >> 5;
    int lane = threadIdx.x & 31;
    int node = blockIdx.x * 8 + wave;
    if (node >= N_NODES) return;

    float4 v = ((const float4*)(in + (size_t)node * HDIM))[lane];
    float s = v.x + v.y + v.z + v.w;
#pragma unroll
    for (int off = 16; off > 0; off >>= 1) s += __shfl_xor(s, off, 32);
    float mu = s * (1.0f / HDIM);

    float4 dd = {v.x - mu, v.y - mu, v.z - mu, v.w - mu};
    float q = dd.x * dd.x + dd.y * dd.y + dd.z * dd.z + dd.w * dd.w;
#pragma unroll
    for (int off = 16; off > 0; off >>= 1) q += __shfl_xor(q, off, 32);
    float rinv = rsqrtf(q * (1.0f / HDIM) + LN_EPS);

    float4 g = ((const float4*)gamma)[lane];
    float4 b = ((const float4*)beta)[lane];
    float4 o;
    o.x = fmaxf(dd.x * rinv * g.x + b.x, 0.0f);
    o.y = fmaxf(dd.y * rinv * g.y + b.y, 0.0f);
    o.z = fmaxf(dd.z * rinv * g.z + b.z, 0.0f);
    o.w = fmaxf(dd.w * rinv * g.w + b.w, 0.0f);
    ((float4*)(out + (size_t)node * HDIM))[lane] = o;
}

// ------------------------------------------------------------------
// Mean pool per graph
// ------------------------------------------------------------------
__global__ void pool_zero_kernel(float* __restrict__ psum, float* __restrict__ pcnt) {
    int i = blockIdx.x * blockDim.x + threadIdx.x;
    if (i < NGRAPH * HDIM) psum[i] = 0.0f;
    if (i < NGRAPH) pcnt[i] = 0.0f;
}

__global__ __launch_bounds__(256)
void pool_acc_kernel(const long long* __restrict__ batch,
                     const float* __restrict__ h,
                     float* __restrict__ psum, float* __restrict__ pcnt) {
    int wave = threadIdx.x >> 5;
    int lane = threadIdx.x & 31;
    int node = blockIdx.x * 8 + wave;
    if (node >= N_NODES) return;
    int g = (int)batch[node];
    float4 v = ((const float4*)(h + (size_t)node * HDIM))[lane];
    float* dst = psum + (size_t)g * HDIM + lane * 4;
    atomicAdd(dst + 0, v.x);
    atomicAdd(dst + 1, v.y);
    atomicAdd(dst + 2, v.z);
    atomicAdd(dst + 3, v.w);
    if (lane == 0) atomicAdd(&pcnt[g], 1.0f);
}

__global__ void pool_div_kernel(float* __restrict__ psum,
                                const float* __restrict__ pcnt) {
    int i = blockIdx.x * blockDim.x + threadIdx.x;
    if (i < NGRAPH * HDIM) {
        float c = pcnt[i >> 7];
        psum[i] = psum[i] / fmaxf(c, 1.0f);
    }
}

// ------------------------------------------------------------------
// Host-side launch
// ------------------------------------------------------------------
extern "C" void kernel_launch(void* const* d_in, const int* in_sizes, int n_in,
                              void* d_out, int out_size, void* d_ws, size_t ws_size,
                              hipStream_t stream) {
    const float*     x        = (const float*)d_in[0];
    const long long* ei       = (const long long*)d_in[1];   // [2, E] int64
    const long long* batch    = (const long long*)d_in[2];   // [N] int64
    const float*     w_in     = (const float*)d_in[3];
    const float*     b_in     = (const float*)d_in[4];
    const float*     w_layers = (const float*)d_in[5];       // [L, H, H]
    const float*     b_layers = (const float*)d_in[6];       // [L, H]
    const float*     gamma    = (const float*)d_in[7];       // [L, H]
    const float*     beta     = (const float*)d_in[8];       // [L, H]
    const float*     w_out    = (const float*)d_in[9];
    const float*     b_out    = (const float*)d_in[10];
    float*           out      = (float*)d_out;

    // Carve workspace (all 512B aligned)
    char* ws = (char*)d_ws;
    size_t off = 0;
    auto alloc = [&](size_t bytes) {
        char* p = ws + off;
        off += (bytes + 511) & ~(size_t)511;
        return p;
    };
    float* deg   = (float*)alloc((size_t)N_NODES * 4);                 // reused as dinv
    float* enorm = (float*)alloc((size_t)(N_EDGES + N_NODES) * 4);
    float* h     = (float*)alloc((size_t)N_NODES * HDIM * 4);
    float* m     = (float*)alloc((size_t)N_NODES * HDIM * 4);
    float* agg   = (float*)alloc((size_t)N_NODES * HDIM * 4);
    float* psum  = (float*)alloc((size_t)NGRAPH * HDIM * 4);
    float* pcnt  = (float*)alloc((size_t)NGRAPH * 4);

    const int T = 256;
    int gN   = (N_NODES + T - 1) / T;
    int gE   = (N_EDGES + T - 1) / T;
    int gEN  = (N_EDGES + N_NODES + T - 1) / T;
    int gEw  = (N_EDGES + N_NODES + 7) / 8;      // wave-per-edge
    int gNw  = (N_NODES + 7) / 8;                // wave-per-node
    int gNH4 = (N_NODES * (HDIM / 4) + T - 1) / T;
    int gGemm = (N_NODES + 127) / 128;

    // GCN normalization (computed once; invariant across layers)
    deg_init_kernel<<<gN, T, 0, stream>>>(deg);
    deg_acc_kernel<<<gE, T, 0, stream>>>(ei, deg);
    dinv_kernel<<<gN, T, 0, stream>>>(deg);
    enorm_kernel<<<gEN, T, 0, stream>>>(ei, deg, enorm);

    // Input projection: h = x @ w_in + b_in
    gemm128_kernel<<<gGemm, T, 0, stream>>>(x, w_in, b_in, h, N_NODES);

    // 8 GCN layers
    for (int l = 0; l < NLAYERS; ++l) {
        gemm128_kernel<<<gGemm, T, 0, stream>>>(h, w_layers + (size_t)l * HDIM * HDIM,
                                                nullptr, m, N_NODES);
        agg_init_kernel<<<gNH4, T, 0, stream>>>(b_layers + (size_t)l * HDIM, agg);
        scatter_kernel<<<gEw, T, 0, stream>>>(ei, enorm, m, agg);
        ln_relu_kernel<<<gNw, T, 0, stream>>>(agg, gamma + (size_t)l * HDIM,
                                              beta + (size_t)l * HDIM, h);
    }

    // Global mean pool + output projection
    pool_zero_kernel<<<(NGRAPH * HDIM + T - 1) / T, T, 0, stream>>>(psum, pcnt);
    pool_acc_kernel<<<gNw, T, 0, stream>>>(batch, h, psum, pcnt);
    pool_div_kernel<<<(NGRAPH * HDIM + T - 1) / T, T, 0, stream>>>(psum, pcnt);
    gemm128_kernel<<<1, T, 0, stream>>>(psum, w_out, b_out, out, NGRAPH);
}